// AdaptiveLogits_14422500180149
// MI455X (gfx1250) — compile-verified
//
#include <hip/hip_runtime.h>

typedef __attribute__((ext_vector_type(2))) float v2f;
typedef __attribute__((ext_vector_type(8))) float v8f;

#define BM 128
#define BN 64
#define BK 32
#define LDSP (BK + 4)
#define NTHREADS 256

static __device__ __forceinline__ v8f wmma_f32(v2f a, v2f b, v8f c) {
  // D = A(16x4) * B(4x16) + C(16x16), all fp32
  return __builtin_amdgcn_wmma_f32_16x16x4_f32(false, a, false, b, (short)0, c, false, false);
}

// ---------------------------------------------------------------------------
// Ordered partition of targets into cluster index lists (single block scan).
// ---------------------------------------------------------------------------
__global__ __launch_bounds__(1024)
void partition_kernel(const int* __restrict__ targets, int n,
                      int* __restrict__ idx0, int* __restrict__ idx1,
                      int* __restrict__ counts) {
  __shared__ int wsum0[32], wsum1[32];
  __shared__ int base0, base1;
  int tid = threadIdx.x, lane = tid & 31, wv = tid >> 5;
  int nw = blockDim.x >> 5;
  if (tid == 0) { base0 = 0; base1 = 0; }
  __syncthreads();
  for (int chunk = 0; chunk < n; chunk += blockDim.x) {
    int i = chunk + tid;
    int t = (i < n) ? targets[i] : -1;
    bool c0 = (t >= 2000) && (t < 20000);
    bool c1 = (t >= 20000);
    unsigned m0 = (unsigned)__ballot(c0);
    unsigned m1 = (unsigned)__ballot(c1);
    if (lane == 0) { wsum0[wv] = __popc(m0); wsum1[wv] = __popc(m1); }
    __syncthreads();
    int off0 = base0, off1 = base1;
    for (int w = 0; w < wv; ++w) { off0 += wsum0[w]; off1 += wsum1[w]; }
    unsigned lt = (lane == 0) ? 0u : (0xFFFFFFFFu >> (32 - lane));
    if (c0) idx0[off0 + __popc(m0 & lt)] = i;
    if (c1) idx1[off1 + __popc(m1 & lt)] = i;
    __syncthreads();
    if (tid == 0) {
      int s0 = 0, s1 = 0;
      for (int w = 0; w < nw; ++w) { s0 += wsum0[w]; s1 += wsum1[w]; }
      base0 += s0; base1 += s1;
    }
    __syncthreads();
  }
  if (tid == 0) { counts[0] = base0; counts[1] = base1; }
}

// ---------------------------------------------------------------------------
// Generic fp32 WMMA GEMM: C[coff + m*ldc + n] = sum_k A'[m][k] * B'[k][n] (+bias)
//   AT: A'[m][k] = A[k*lda+m]   else A'[m][k] = A[row(m)*lda+k]
//   GA: row(m) = idx[m]         else row(m) = m
//   BT: B'[k][n] = B[n*ldb+k]   else B'[k][n] = B[k*ldb+n]
//   Effective M read from *mptr (if non-null); coff = (*coff_cnt)*coff_scale.
// ---------------------------------------------------------------------------
template<bool AT, bool BT, bool GA, bool HB>
__global__ __launch_bounds__(NTHREADS)
void gemm_wmma_f32(const float* __restrict__ A, int lda,
                   const float* __restrict__ B, int ldb,
                   float* __restrict__ C, long long ldc,
                   int Mmax, const int* __restrict__ mptr,
                   int N, int K,
                   const int* __restrict__ idx,
                   const float* __restrict__ bias,
                   const int* __restrict__ coff_cnt, long long coff_scale)
{
  int Meff = mptr ? *mptr : Mmax;
  int mBase = (int)blockIdx.y * BM;
  int nBase = (int)blockIdx.x * BN;
  if (mBase >= Meff) return;
  long long coff = coff_cnt ? (long long)(*coff_cnt) * coff_scale : 0ll;

  __shared__ __align__(16) float As[BM][LDSP];
  __shared__ __align__(16) float Bs[BN][LDSP];

  int tid  = threadIdx.x;
  int lane = tid & 31;
  int wave = tid >> 5;
  int wr = (wave & 3) << 5;   // 4 wave-rows of 32
  int wc = (wave >> 2) << 5;  // 2 wave-cols of 32
  int ml = lane & 15;
  int kh = (lane >> 4) << 1;  // K-half select: 0 or 2

  v8f acc[2][2] = {};

  for (int k0 = 0; k0 < K; k0 += BK) {
    // ---- stage A tile (BM x BK) into LDS, K-contiguous ----
    if (!AT) {
      #pragma unroll
      for (int i = 0; i < (BM * BK) / (NTHREADS * 4); ++i) {
        int linear = tid + i * NTHREADS;
        int row = linear >> 3;            // BK/4 = 8 float4 per row
        int kq  = (linear & 7) << 2;
        float4 v = {0.f, 0.f, 0.f, 0.f};
        int gm = mBase + row;
        if (gm < Meff) {
          long long ar = GA ? (long long)idx[gm] : (long long)gm;
          v = *(const float4*)(A + ar * (long long)lda + (k0 + kq));
        }
        *(float4*)&As[row][kq] = v;
      }
    } else {
      #pragma unroll
      for (int i = 0; i < (BM * BK) / (NTHREADS * 4); ++i) {
        int linear = tid + i * NTHREADS;
        int kr = linear >> 5;             // BM/4 = 32 float4 per k-row
        int m4 = (linear & 31) << 2;
        int gm = mBase + m4;
        float4 v = {0.f, 0.f, 0.f, 0.f};
        if (gm + 3 < Meff)
          v = *(const float4*)(A + (long long)(k0 + kr) * lda + gm);
        As[m4 + 0][kr] = v.x; As[m4 + 1][kr] = v.y;
        As[m4 + 2][kr] = v.z; As[m4 + 3][kr] = v.w;
      }
    }
    // ---- stage B tile (BK x BN) into LDS as [n][k] ----
    if (BT) {
      #pragma unroll
      for (int i = 0; i < (BN * BK) / (NTHREADS * 4); ++i) {
        int linear = tid + i * NTHREADS;
        int n  = linear >> 3;
        int kq = (linear & 7) << 2;
        float4 v = {0.f, 0.f, 0.f, 0.f};
        int gn = nBase + n;
        if (gn < N)
          v = *(const float4*)(B + (long long)gn * ldb + (k0 + kq));
        *(float4*)&Bs[n][kq] = v;
      }
    } else {
      #pragma unroll
      for (int i = 0; i < (BN * BK) / (NTHREADS * 4); ++i) {
        int linear = tid + i * NTHREADS;
        int kr = linear >> 4;             // BN/4 = 16 float4 per k-row
        int n4 = (linear & 15) << 2;
        int gn = nBase + n4;
        float4 v = {0.f, 0.f, 0.f, 0.f};
        if (gn + 3 < N)
          v = *(const float4*)(B + (long long)(k0 + kr) * ldb + gn);
        Bs[n4 + 0][kr] = v.x; Bs[n4 + 1][kr] = v.y;
        Bs[n4 + 2][kr] = v.z; Bs[n4 + 3][kr] = v.w;
      }
    }
    __syncthreads();

    // ---- 8 K-steps of 4, 2x2 wave subtiles -> 32 WMMAs per stage ----
    #pragma unroll
    for (int kk = 0; kk < BK; kk += 4) {
      int ks = kk + kh;
      v2f a0 = *(const v2f*)&As[wr + ml][ks];
      v2f a1 = *(const v2f*)&As[wr + 16 + ml][ks];
      v2f b0 = *(const v2f*)&Bs[wc + ml][ks];
      v2f b1 = *(const v2f*)&Bs[wc + 16 + ml][ks];
      acc[0][0] = wmma_f32(a0, b0, acc[0][0]);
      acc[0][1] = wmma_f32(a0, b1, acc[0][1]);
      acc[1][0] = wmma_f32(a1, b0, acc[1][0]);
      acc[1][1] = wmma_f32(a1, b1, acc[1][1]);
    }
    __syncthreads();
  }

  // ---- epilogue: C layout VGPR r -> M = r + 8*(lane>=16), N = lane&15 ----
  int rsel = (lane >> 4) << 3;
  #pragma unroll
  for (int i = 0; i < 2; ++i) {
    #pragma unroll
    for (int j = 0; j < 2; ++j) {
      int gn  = nBase + wc + j * 16 + ml;
      int gmb = mBase + wr + i * 16 + rsel;
      #pragma unroll
      for (int r = 0; r < 8; ++r) {
        int gm = gmb + r;
        if (gm < Meff && gn < N) {
          float v = acc[i][j][r];
          if (HB) v += bias[gn];
          C[coff + (long long)gm * ldc + gn] = v;
        }
      }
    }
  }
}

// ---------------------------------------------------------------------------
extern "C" void kernel_launch(void* const* d_in, const int* in_sizes, int n_in,
                              void* d_out, int out_size, void* d_ws, size_t ws_size,
                              hipStream_t stream) {
  (void)in_sizes; (void)n_in; (void)out_size; (void)ws_size;
  const float* hidden = (const float*)d_in[0];
  const float* embed  = (const float*)d_in[1];
  const float* tailW  = (const float*)d_in[2];
  const float* tailB  = (const float*)d_in[3];
  const float* slBias = (const float*)d_in[4];
  const float* bias0  = (const float*)d_in[5];
  const float* bias1  = (const float*)d_in[6];
  const float* down0  = (const float*)d_in[7];
  const float* down1  = (const float*)d_in[8];
  const int*   targets= (const int*)d_in[9];
  float* out = (float*)d_out;

  const int Bb = 8192, H = 1024, C0 = 2000, C1 = 20000, C2 = 50000;
  const int N0v = C1 - C0;          // 18000
  const int N1v = C2 - C1;          // 30000
  const int Hq  = H / 4;            // 256
  const long long HEAD = (long long)Bb * (C0 + 2);

  // workspace layout
  int*   wsI   = (int*)d_ws;
  int*   counts= wsI;                        // [0]=n0 [1]=n1
  int*   idx0  = wsI + 16;
  int*   idx1  = wsI + 16 + Bb;
  float* wsF   = (float*)d_ws + 32768;       // byte 131072
  float* G0    = wsF;                        // 1024*1024
  float* dec1  = G0   + (long long)H * H;    // 30000*256
  float* proj0 = dec1 + (long long)N1v * Hq; // 8192*1024 (max)
  float* proj1 = proj0 + (long long)Bb * H;  // 8192*256 (max)

  // 1) ordered cluster partition
  partition_kernel<<<1, 1024, 0, stream>>>(targets, Bb, idx0, idx1, counts);

  // 2) head = hidden @ shortlistT + slBias  (N=2000, ldc=2002)
  gemm_wmma_f32<false, true, false, true>
    <<<dim3((C0 + BN - 1) / BN, Bb / BM), NTHREADS, 0, stream>>>(
      hidden, H, embed, H, out, (long long)(C0 + 2),
      Bb, nullptr, C0, H, nullptr, slBias, nullptr, 0);

  // 3) head tail cols 2000..2001 = hidden @ tailWT + tailB
  gemm_wmma_f32<false, true, false, true>
    <<<dim3(1, Bb / BM), NTHREADS, 0, stream>>>(
      hidden, H, tailW, H, out + C0, (long long)(C0 + 2),
      Bb, nullptr, 2, H, nullptr, tailB, nullptr, 0);

  // 4) G0 = down0T @ down0 (1024x1024)
  gemm_wmma_f32<true, false, false, false>
    <<<dim3(H / BN, H / BM), NTHREADS, 0, stream>>>(
      down0, H, down0, H, G0, (long long)H,
      H, nullptr, H, H, nullptr, nullptr, nullptr, 0);

  // 5) dec1 = embed1 @ down1T (30000x256)
  gemm_wmma_f32<false, true, false, false>
    <<<dim3(Hq / BN, (N1v + BM - 1) / BM), NTHREADS, 0, stream>>>(
      embed + (long long)C1 * H, H, down1, H, dec1, (long long)Hq,
      N1v, nullptr, Hq, H, nullptr, nullptr, nullptr, 0);

  // 6) proj0 = hidden[idx0] @ G0 (n0 x 1024)
  gemm_wmma_f32<false, false, true, false>
    <<<dim3(H / BN, Bb / BM), NTHREADS, 0, stream>>>(
      hidden, H, G0, H, proj0, (long long)H,
      Bb, counts + 0, H, H, idx0, nullptr, nullptr, 0);

  // 7) proj1 = hidden[idx1] @ down1T (n1 x 256)
  gemm_wmma_f32<false, true, true, false>
    <<<dim3(Hq / BN, Bb / BM), NTHREADS, 0, stream>>>(
      hidden, H, down1, H, proj1, (long long)Hq,
      Bb, counts + 1, Hq, H, idx1, nullptr, nullptr, 0);

  // 8) out0 = proj0 @ embed0T + bias0 (n0 x 18000) at d_out+HEAD
  gemm_wmma_f32<false, true, false, true>
    <<<dim3((N0v + BN - 1) / BN, Bb / BM), NTHREADS, 0, stream>>>(
      proj0, H, embed + (long long)C0 * H, H, out + HEAD, (long long)N0v,
      Bb, counts + 0, N0v, H, nullptr, bias0, nullptr, 0);

  // 9) out1 = proj1 @ dec1T + bias1 (n1 x 30000) at d_out+HEAD+n0*18000
  gemm_wmma_f32<false, true, false, true>
    <<<dim3((N1v + BN - 1) / BN, Bb / BM), NTHREADS, 0, stream>>>(
      proj1, Hq, dec1, Hq, out + HEAD, (long long)N1v,
      Bb, counts + 1, N1v, Hq, nullptr, bias1, counts + 0, (long long)N0v);
}